// Homo_77051713290671
// MI455X (gfx1250) — compile-verified
//
#include <hip/hip_runtime.h>
#include <math.h>

typedef float v2f __attribute__((ext_vector_type(2)));
typedef float v8f __attribute__((ext_vector_type(8)));

#define ND 5000
#define NP 15000
#define NNODE 20000     // Nd + Np
#define FEAT 256
#define EPOS 250000
#define M_EDGES 500000  // 2E
#define H0 128
#define H1 64
#define H2 32

__device__ __forceinline__ const float* node_row(const float* drug, const float* prot, int idx) {
    return (idx < ND) ? (drug + (size_t)idx * FEAT)
                      : (prot + (size_t)(idx - ND) * FEAT);
}

__device__ __forceinline__ void edge_sd(const int* pos, const int* neg, int e, int& s, int& d) {
    if (e < EPOS) { s = pos[e];        d = pos[EPOS + e]; }
    else          { int f = e - EPOS; s = neg[f]; d = neg[EPOS + f]; }
}

// fp32 WMMA: D = A(16x4) * B(4x16) + C  (16x16 f32 accum)
__device__ __forceinline__ v8f wmma_f32(v2f a, v2f b, v8f c) {
    return __builtin_amdgcn_wmma_f32_16x16x4_f32(false, a, false, b, (short)0, c, false, false);
}

// ---------------------------------------------------------------------------
// K1: xh[i] = n_fea[src[i]] * n_fea[dst[i]], i in [0, 20000).  float4 path.
// grid = NNODE, block = FEAT/4
// ---------------------------------------------------------------------------
__global__ void xh_kernel(const float* __restrict__ drug, const float* __restrict__ prot,
                          const int* __restrict__ pos, float* __restrict__ xh) {
    int i = blockIdx.x;
    int s = pos[i], d = pos[EPOS + i];
    const float4* fs = (const float4*)node_row(drug, prot, s);
    const float4* fd = (const float4*)node_row(drug, prot, d);
    int c = threadIdx.x;                 // 0..63
    float4 a = fs[c], b = fd[c];
    float4 p = make_float4(a.x * b.x, a.y * b.y, a.z * b.z, a.w * b.w);
    ((float4*)(xh + (size_t)i * FEAT))[c] = p;
}

// ---------------------------------------------------------------------------
// K2: y = xh[20000x256] @ Wl[256x128].  One wave per 16x16 tile.
// grid = (1250, 8), block = 32
// ---------------------------------------------------------------------------
__global__ void gemm_y_kernel(const float* __restrict__ xh, const float* __restrict__ Wl,
                              float* __restrict__ y) {
    const int lane = threadIdx.x & 31;
    const int half = lane >> 4, lq = lane & 15;
    const int r0 = blockIdx.x * 16;
    const int c0 = blockIdx.y * 16;
    v8f acc = {};
    for (int k = 0; k < FEAT; k += 4) {
        v2f a, b;
        const float* ap = xh + (size_t)(r0 + lq) * FEAT + k + 2 * half;
        a.x = ap[0]; a.y = ap[1];
        const float* bp = Wl + (size_t)(k + 2 * half) * H0 + c0 + lq;
        b.x = bp[0]; b.y = bp[H0];
        acc = wmma_f32(a, b, acc);
    }
    for (int j = 0; j < 8; ++j) {
        int m = j + 8 * half;
        y[(size_t)(r0 + m) * H0 + c0 + lq] = acc[j];
    }
}

// ---------------------------------------------------------------------------
// K3: zero z[20000x128] and degf[20000] (atomics accumulate each call)
// ---------------------------------------------------------------------------
__global__ void zero_kernel(float* __restrict__ z, float* __restrict__ degf) {
    int i = blockIdx.x * blockDim.x + threadIdx.x;
    if (i < NNODE * H0) z[i] = 0.0f;
    if (i < NNODE) degf[i] = 0.0f;
}

// ---------------------------------------------------------------------------
// K4: scatter-add in projected 128-dim space:
//     z[dst[e]] += y[src[e]] ; degf[dst[e]] += 1
// 2 edges per 256-thread block; grid = M_EDGES/2
// ---------------------------------------------------------------------------
__global__ void scatter_kernel(const float* __restrict__ y,
                               const int* __restrict__ pos, const int* __restrict__ neg,
                               float* __restrict__ z, float* __restrict__ degf) {
    int e = blockIdx.x * 2 + (threadIdx.x >> 7);
    int c = threadIdx.x & (H0 - 1);
    int s, d;
    edge_sd(pos, neg, e, s, d);
    atomicAdd(&z[(size_t)d * H0 + c], y[(size_t)s * H0 + c]);
    if (c == 0) atomicAdd(&degf[d], 1.0f);
}

// ---------------------------------------------------------------------------
// K5: fused  x -> (x@Wr + bl + z/deg) -> relu -> @W1 -> relu -> @W2 -> relu
//            -> @W3 -> log_softmax -> prob, label
// 32 edges per block, 256 threads (8 waves)
// ---------------------------------------------------------------------------
__global__ void __launch_bounds__(256)
fused_kernel(const float* __restrict__ drug, const float* __restrict__ prot,
             const int* __restrict__ pos, const int* __restrict__ neg,
             const float* __restrict__ Wr, const float* __restrict__ bl,
             const float* __restrict__ z, const float* __restrict__ degf,
             const float* __restrict__ W1, const float* __restrict__ b1,
             const float* __restrict__ W2, const float* __restrict__ b2,
             const float* __restrict__ W3, const float* __restrict__ b3,
             float* __restrict__ prob, float* __restrict__ label) {
    __shared__ float xbuf[32 * FEAT];   // 32 KB; reused as h0[32*128] after layer0
    __shared__ float h1buf[32 * H1];    // 8 KB
    __shared__ float h2buf[32 * H2];    // 4 KB
    __shared__ int s_idx[32], d_idx[32];

    const int t    = threadIdx.x;
    const int lane = t & 31;
    const int w    = t >> 5;            // wave id 0..7
    const int half = lane >> 4, lq = lane & 15;
    const int eb   = blockIdx.x * 32;

    // ---- phase 0: edge indices + x tile into LDS (float4 / b128 path)
    if (t < 32) {
        int s, d;
        edge_sd(pos, neg, eb + t, s, d);
        s_idx[t] = s; d_idx[t] = d;
    }
    __syncthreads();
    {
        float4* xb4 = (float4*)xbuf;
        for (int i = t; i < 32 * (FEAT / 4); i += 256) {
            int r = i >> 6, c4 = i & 63;
            const float4* fs = (const float4*)node_row(drug, prot, s_idx[r]);
            const float4* fd = (const float4*)node_row(drug, prot, d_idx[r]);
            float4 a = fs[c4], b = fd[c4];
            xb4[i] = make_float4(a.x * b.x, a.y * b.y, a.z * b.z, a.w * b.w);
        }
    }
    __syncthreads();

    // ---- phase 1: h0 = relu(x @ Wr + bl + z/deg); wave w owns cols [16w,16w+16)
    {
        v8f c0 = {}, c1 = {};
        const int n = 16 * w + lq;
        for (int k = 0; k < FEAT; k += 4) {
            v2f b;
            b.x = Wr[(size_t)(k + 2 * half) * H0 + n];
            b.y = Wr[(size_t)(k + 2 * half + 1) * H0 + n];
            v2f a0, a1;
            const float* p0 = &xbuf[lq * FEAT + k + 2 * half];
            const float* p1 = &xbuf[(16 + lq) * FEAT + k + 2 * half];
            a0.x = p0[0]; a0.y = p0[1];
            a1.x = p1[0]; a1.y = p1[1];
            c0 = wmma_f32(a0, b, c0);
            c1 = wmma_f32(a1, b, c1);
        }
        __syncthreads();                 // all x reads done; xbuf now becomes h0
        float* h0buf = xbuf;
        const float blv = bl[n];
        for (int rt = 0; rt < 2; ++rt) {
            v8f acc = rt ? c1 : c0;
            for (int j = 0; j < 8; ++j) {
                int row = rt * 16 + j + 8 * half;
                float v = acc[j] + blv;
                int e = eb + row;
                if (e < NNODE)           // mean-aggregation term (zero for e>=N)
                    v += z[(size_t)e * H0 + n] / fmaxf(degf[e], 1.0f);
                h0buf[row * H0 + n] = fmaxf(v, 0.0f);
            }
        }
        __syncthreads();
    }

    // ---- phase 2: h1 = relu(h0[32x128] @ W1[128x64] + b1); 8 tiles, 1/wave
    {
        float* h0buf = xbuf;
        const int rt = w >> 2, cb = w & 3;
        const int n = 16 * cb + lq;
        v8f acc = {};
        for (int k = 0; k < H0; k += 4) {
            v2f a, b;
            const float* ap = &h0buf[(rt * 16 + lq) * H0 + k + 2 * half];
            a.x = ap[0]; a.y = ap[1];
            b.x = W1[(size_t)(k + 2 * half) * H1 + n];
            b.y = W1[(size_t)(k + 2 * half + 1) * H1 + n];
            acc = wmma_f32(a, b, acc);
        }
        const float b1v = b1[n];
        for (int j = 0; j < 8; ++j) {
            int row = rt * 16 + j + 8 * half;
            h1buf[row * H1 + n] = fmaxf(acc[j] + b1v, 0.0f);
        }
        __syncthreads();
    }

    // ---- phase 3: h2 = relu(h1[32x64] @ W2[64x32] + b2); 4 tiles, waves 0-3
    if (w < 4) {
        const int rt = w >> 1, cb = w & 1;
        const int n = 16 * cb + lq;
        v8f acc = {};
        for (int k = 0; k < H1; k += 4) {
            v2f a, b;
            const float* ap = &h1buf[(rt * 16 + lq) * H1 + k + 2 * half];
            a.x = ap[0]; a.y = ap[1];
            b.x = W2[(size_t)(k + 2 * half) * H2 + n];
            b.y = W2[(size_t)(k + 2 * half + 1) * H2 + n];
            acc = wmma_f32(a, b, acc);
        }
        const float b2v = b2[n];
        for (int j = 0; j < 8; ++j) {
            int row = rt * 16 + j + 8 * half;
            h2buf[row * H2 + n] = fmaxf(acc[j] + b2v, 0.0f);
        }
    }
    __syncthreads();

    // ---- phase 4: logits (32x32 @ 32x2 + b3), log_softmax, outputs
    if (t < 32) {
        float l0 = b3[0], l1 = b3[1];
        for (int k = 0; k < H2; ++k) {
            float hv = h2buf[t * H2 + k];
            l0 += hv * W3[k * 2 + 0];
            l1 += hv * W3[k * 2 + 1];
        }
        float mx  = fmaxf(l0, l1);
        float lse = mx + logf(expf(l0 - mx) + expf(l1 - mx));
        int e = eb + t;
        prob[(size_t)e * 2 + 0] = l0 - lse;
        prob[(size_t)e * 2 + 1] = l1 - lse;
        float lab = (e < EPOS) ? 1.0f : 0.0f;
        label[(size_t)e * 2 + 0] = lab;
        label[(size_t)e * 2 + 1] = lab;
    }
}

// ---------------------------------------------------------------------------
extern "C" void kernel_launch(void* const* d_in, const int* in_sizes, int n_in,
                              void* d_out, int out_size, void* d_ws, size_t ws_size,
                              hipStream_t stream) {
    (void)in_sizes; (void)n_in; (void)out_size; (void)ws_size;
    const float* drug = (const float*)d_in[0];
    const float* prot = (const float*)d_in[1];
    const float* Wl   = (const float*)d_in[2];
    const float* bl   = (const float*)d_in[3];
    const float* Wr   = (const float*)d_in[4];
    const float* W1   = (const float*)d_in[5];
    const float* b1   = (const float*)d_in[6];
    const float* W2   = (const float*)d_in[7];
    const float* b2   = (const float*)d_in[8];
    const float* W3   = (const float*)d_in[9];
    const float* b3   = (const float*)d_in[10];
    const int*   pos  = (const int*)d_in[11];
    const int*   neg  = (const int*)d_in[12];

    float* prob  = (float*)d_out;
    float* label = prob + (size_t)M_EDGES * 2;

    float* xh   = (float*)d_ws;                    // 20000*256
    float* y    = xh + (size_t)NNODE * FEAT;       // 20000*128
    float* z    = y  + (size_t)NNODE * H0;         // 20000*128
    float* degf = z  + (size_t)NNODE * H0;         // 20000

    xh_kernel<<<NNODE, FEAT / 4, 0, stream>>>(drug, prot, pos, xh);
    gemm_y_kernel<<<dim3(NNODE / 16, H0 / 16), 32, 0, stream>>>(xh, Wl, y);
    zero_kernel<<<(NNODE * H0 + 255) / 256, 256, 0, stream>>>(z, degf);
    scatter_kernel<<<M_EDGES / 2, 256, 0, stream>>>(y, pos, neg, z, degf);
    fused_kernel<<<M_EDGES / 32, 256, 0, stream>>>(drug, prot, pos, neg,
                                                   Wr, bl, z, degf,
                                                   W1, b1, W2, b2, W3, b3,
                                                   prob, label);
}